// ClosePenalty_59304908423819
// MI455X (gfx1250) — compile-verified
//
#include <hip/hip_runtime.h>
#include <stdint.h>

#define N_ATM   10000
#define N_BIN   8
#define ELM_MAX 128      // table is 100 entries; round up for LDS
#define BLK     256      // 8 wave32 per block
#define NBLOCKS 1280

#if defined(__gfx1250__) && __has_builtin(__builtin_amdgcn_global_load_async_to_lds_b32) && __has_builtin(__builtin_amdgcn_s_wait_asynccnt)
#define HAVE_ASYNC_LDS 1
#else
#define HAVE_ASYNC_LDS 0
#endif

__global__ __launch_bounds__(BLK) void edge_energy_kernel(
    const int*   __restrict__ elm,     // [N_BCH * N_ATM]
    const int*   __restrict__ edge_n,  // [nEdge]
    const int*   __restrict__ edge_i,  // [nEdge]
    const int*   __restrict__ edge_j,  // [nEdge]
    const float* __restrict__ sod,     // [nEdge]
    const float* __restrict__ kp,      // [nElm]
    const float* __restrict__ rad,     // [nElm]
    int nElm, int nEdge,
    float* __restrict__ partials)      // [gridDim.x * 8]
{
    __shared__ float2 kr[ELM_MAX];          // packed {k, radius} table
    __shared__ float  wsum[BLK / 32][N_BIN];

    const int tid = threadIdx.x;

    // ---- stage k/radius table into LDS (async-to-LDS path if available) ----
#if HAVE_ASYNC_LDS
    if (tid < nElm) {
        typedef int __attribute__((address_space(1))) gint;
        typedef int __attribute__((address_space(3))) lint;
        gint* gk = (gint*)(kp  + tid);
        gint* gr = (gint*)(rad + tid);
        lint* lk = (lint*)&kr[tid].x;
        lint* lr = (lint*)&kr[tid].y;
        __builtin_amdgcn_global_load_async_to_lds_b32(gk, lk, 0, 0);
        __builtin_amdgcn_global_load_async_to_lds_b32(gr, lr, 0, 0);
    }
    __builtin_amdgcn_s_wait_asynccnt(0);
#else
    if (tid < nElm) {
        kr[tid] = make_float2(kp[tid], rad[tid]);
    }
#endif
    __syncthreads();

    float acc[N_BIN];
#pragma unroll
    for (int b = 0; b < N_BIN; ++b) acc[b] = 0.0f;

    auto process = [&](int n, int i, int j, float s) {
        const int base = n * N_ATM;
        const int ti = elm[base + i];          // L2-resident gather (320 KB)
        const int tj = elm[base + j];
        const float2 a = kr[ti];               // LDS, 64-bit load
        const float2 b = kr[tj];
        const float R   = a.y + b.y;
        const float dis = __builtin_amdgcn_sqrtf(s);
        const float d   = dis - R;
        float en = (a.x + b.x) * d * d;
        en = (dis < R) ? en : 0.0f;
#pragma unroll
        for (int bn = 0; bn < N_BIN; ++bn)     // select into 8 register bins
            acc[bn] += (n == bn) ? en : 0.0f;
    };

    const int nQuad   = nEdge >> 2;
    const int gstride = (int)(gridDim.x * blockDim.x);
    const int4*   en4  = (const int4*)edge_n;
    const int4*   ei4  = (const int4*)edge_i;
    const int4*   ej4  = (const int4*)edge_j;
    const float4* sod4 = (const float4*)sod;

    for (int q = (int)(blockIdx.x * blockDim.x) + tid; q < nQuad; q += gstride) {
        // speculative prefetch of next stride -> global_prefetch_b8
        __builtin_prefetch(&en4[q + gstride],  0, 0);
        __builtin_prefetch(&ei4[q + gstride],  0, 0);
        __builtin_prefetch(&ej4[q + gstride],  0, 0);
        __builtin_prefetch(&sod4[q + gstride], 0, 0);

        const int4   n4 = en4[q];
        const int4   i4 = ei4[q];
        const int4   j4 = ej4[q];
        const float4 s4 = sod4[q];
        process(n4.x, i4.x, j4.x, s4.x);
        process(n4.y, i4.y, j4.y, s4.y);
        process(n4.z, i4.z, j4.z, s4.z);
        process(n4.w, i4.w, j4.w, s4.w);
    }

    // scalar tail (nEdge not multiple of 4)
    for (int e = (nQuad << 2) + (int)(blockIdx.x * blockDim.x) + tid; e < nEdge; e += gstride)
        process(edge_n[e], edge_i[e], edge_j[e], sod[e]);

    // ---- wave32 butterfly reduction, then cross-wave via LDS ----
    const int lane = tid & 31;
    const int wave = tid >> 5;
#pragma unroll
    for (int bn = 0; bn < N_BIN; ++bn) {
        float v = acc[bn];
#pragma unroll
        for (int off = 16; off > 0; off >>= 1)
            v += __shfl_xor(v, off, 32);
        if (lane == 0) wsum[wave][bn] = v;
    }
    __syncthreads();
    if (tid < N_BIN) {
        float s = 0.0f;
#pragma unroll
        for (int w = 0; w < BLK / 32; ++w) s += wsum[w][tid];
        partials[blockIdx.x * N_BIN + tid] = s;
    }
}

__global__ __launch_bounds__(BLK) void reduce_partials_kernel(
    const float* __restrict__ partials, int nPart, float* __restrict__ out)
{
    __shared__ float sm[32][N_BIN];
    const int tid = threadIdx.x;
    const int bin = tid & 7;
    const int r   = tid >> 3;   // 0..31
    float s = 0.0f;
    for (int i = r; i < nPart; i += 32)
        s += partials[i * N_BIN + bin];
    sm[r][bin] = s;
    __syncthreads();
    if (tid < N_BIN) {
        float t = 0.0f;
#pragma unroll
        for (int i = 0; i < 32; ++i) t += sm[i][tid];
        out[tid] = t;
    }
}

extern "C" void kernel_launch(void* const* d_in, const int* in_sizes, int n_in,
                              void* d_out, int out_size, void* d_ws, size_t ws_size,
                              hipStream_t stream) {
    const int*   elm    = (const int*)d_in[0];
    const int*   edge_n = (const int*)d_in[1];
    const int*   edge_i = (const int*)d_in[2];
    const int*   edge_j = (const int*)d_in[3];
    const float* sod    = (const float*)d_in[4];
    const float* kp     = (const float*)d_in[5];
    const float* rad    = (const float*)d_in[6];
    const int nEdge = in_sizes[1];
    const int nElm  = in_sizes[5];

    float* out      = (float*)d_out;
    float* partials = (float*)d_ws;

    int nblocks = NBLOCKS;
    const size_t need = (size_t)nblocks * N_BIN * sizeof(float);
    if (need > ws_size) {
        nblocks = (int)(ws_size / (N_BIN * sizeof(float)));
        if (nblocks < 1) nblocks = 1;
    }

    edge_energy_kernel<<<nblocks, BLK, 0, stream>>>(
        elm, edge_n, edge_i, edge_j, sod, kp, rad, nElm, nEdge, partials);
    reduce_partials_kernel<<<1, BLK, 0, stream>>>(partials, nblocks, out);
}